// StructureEncoding_21912923144252
// MI455X (gfx1250) — compile-verified
//
#include <hip/hip_runtime.h>
#include <math.h>

typedef __attribute__((ext_vector_type(2))) float v2f;
typedef __attribute__((ext_vector_type(8))) float v8f;

#define N_NODES 50000
#define N_EDGES 800000
#define IN_DIM  128
#define EDGE_DIM 16
#define EDGE_HID 64
#define NH 8
#define HD 8
#define HDIM 64  /* NH*HD */

// ---- monotone float<->uint encoding so unsigned atomicMax == float max ----
__device__ __forceinline__ unsigned encOrd(float f) {
  unsigned b = __float_as_uint(f);
  return (b & 0x80000000u) ? ~b : (b | 0x80000000u);
}
__device__ __forceinline__ float decOrd(unsigned u) {
  unsigned b = (u & 0x80000000u) ? (u ^ 0x80000000u) : ~u;
  return __uint_as_float(b);
}

// ---------------- kernel 0: tiny weight folding ----------------
// Wc[h*8+k, i] = sum_d W_att[k,d] * W[h*8+d, i]        (64 x 128)
// Wec[h, c]    = sum_m W_edge_att[h,m] * W_edge[m,c]   (8 x 16)
__global__ __launch_bounds__(256) void prep_kernel(
    const float* __restrict__ W, const float* __restrict__ W_att,
    const float* __restrict__ W_edge, const float* __restrict__ W_edge_att,
    float* __restrict__ Wc, float* __restrict__ Wec) {
  for (int idx = threadIdx.x; idx < HDIM * IN_DIM; idx += blockDim.x) {
    int j = idx / IN_DIM, i = idx % IN_DIM;
    int h = j / HD, k = j % HD;
    float acc = 0.f;
    #pragma unroll
    for (int d = 0; d < HD; ++d)
      acc += W_att[k * HD + d] * W[(h * HD + d) * IN_DIM + i];
    Wc[idx] = acc;
  }
  for (int idx = threadIdx.x; idx < NH * EDGE_DIM; idx += blockDim.x) {
    int h = idx / EDGE_DIM, c = idx % EDGE_DIM;
    float acc = 0.f;
    for (int m = 0; m < EDGE_HID; ++m)
      acc += W_edge_att[h * EDGE_HID + m] * W_edge[m * EDGE_DIM + c];
    Wec[idx] = acc;
  }
}

// ---------------- kernel 1: per-call accumulator init ----------------
__global__ __launch_bounds__(256) void init_kernel(unsigned* __restrict__ hmax,
                                                   float* __restrict__ hsum,
                                                   float* __restrict__ sacc) {
  int i = blockIdx.x * blockDim.x + threadIdx.x;
  if (i < NH) { hmax[i] = 0x007FFFFFu /* encOrd(-inf) */; hsum[i] = 0.f; }
  if (i < N_NODES * NH) sacc[i] = 0.f;
}

// ---------------- kernel 2: g = x @ Wc^T via V_WMMA_F32_16X16X4_F32 -------
// One wave computes a 16(M-rows) x 64(all cols) strip. 32 K-steps of 4.
// A layout: lane l -> row (l&15); VGPR v holds K = 2*(l>>4)+v  -> float2 load.
// B layout: lane l -> col (l&15); VGPR v holds K-row v+2*(l>>4) -> float2 load.
// C/D layout: VGPR r -> row r + 8*(l>>4), col (l&15).
__global__ __launch_bounds__(256) void g_gemm_kernel(
    const float* __restrict__ x, const float* __restrict__ Wc,
    float* __restrict__ g) {
  const int wavesPerBlk = blockDim.x >> 5;
  const int wave = blockIdx.x * wavesPerBlk + (threadIdx.x >> 5);
  const int lane = threadIdx.x & 31;
  const int rowBase = wave * 16;
  if (rowBase >= N_NODES) return;  // uniform per wave: EXEC stays all-ones
  const int half = lane >> 4;
  const int l16  = lane & 15;

  v8f acc0 = {}, acc1 = {}, acc2 = {}, acc3 = {};
  const float* xrow = x + (size_t)(rowBase + l16) * IN_DIM;
  const float* b0 = Wc + (size_t)(0 * 16 + l16) * IN_DIM + 2 * half;
  const float* b1 = Wc + (size_t)(1 * 16 + l16) * IN_DIM + 2 * half;
  const float* b2 = Wc + (size_t)(2 * 16 + l16) * IN_DIM + 2 * half;
  const float* b3 = Wc + (size_t)(3 * 16 + l16) * IN_DIM + 2 * half;

  for (int kb = 0; kb < IN_DIM; kb += 4) {
    v2f a  = *(const v2f*)(xrow + kb + 2 * half);
    v2f vb0 = *(const v2f*)(b0 + kb);
    v2f vb1 = *(const v2f*)(b1 + kb);
    v2f vb2 = *(const v2f*)(b2 + kb);
    v2f vb3 = *(const v2f*)(b3 + kb);
    acc0 = __builtin_amdgcn_wmma_f32_16x16x4_f32(false, a, false, vb0, (short)0, acc0, false, false);
    acc1 = __builtin_amdgcn_wmma_f32_16x16x4_f32(false, a, false, vb1, (short)0, acc1, false, false);
    acc2 = __builtin_amdgcn_wmma_f32_16x16x4_f32(false, a, false, vb2, (short)0, acc2, false, false);
    acc3 = __builtin_amdgcn_wmma_f32_16x16x4_f32(false, a, false, vb3, (short)0, acc3, false, false);
  }
  #pragma unroll
  for (int r = 0; r < 8; ++r) {
    int m = r + 8 * half;
    float* grow = g + (size_t)(rowBase + m) * HDIM + l16;
    grow[0]  = acc0[r];
    grow[16] = acc1[r];
    grow[32] = acc2[r];
    grow[48] = acc3[r];
  }
}

// ---------------- kernel 3: raw attention logits + per-head max -----------
__global__ __launch_bounds__(256) void alpha_kernel(
    const float* __restrict__ g, const float* __restrict__ edge_attr,
    const float* __restrict__ Wec, const int* __restrict__ src,
    const int* __restrict__ dst, float* __restrict__ alpha,
    unsigned* __restrict__ hmax) {
  __shared__ float sWec[NH * EDGE_DIM];
  __shared__ unsigned smax[NH];
  const int tid = threadIdx.x;
  if (tid < NH * EDGE_DIM) sWec[tid] = Wec[tid];
  if (tid < NH) smax[tid] = 0x007FFFFFu;  // encOrd(-inf)
  __syncthreads();

  const int e = blockIdx.x * blockDim.x + tid;
  if (e < N_EDGES) {
    const int sI = src[e], dI = dst[e];
    const float* gs = g + (size_t)sI * HDIM;
    const float* gd = g + (size_t)dI * HDIM;
    const float* ea = edge_attr + (size_t)e * EDGE_DIM;
    float eav[EDGE_DIM];
    #pragma unroll
    for (int c = 0; c < EDGE_DIM; ++c) eav[c] = ea[c];
    #pragma unroll
    for (int h = 0; h < NH; ++h) {
      float dotv = 0.f;
      #pragma unroll
      for (int d = 0; d < HD; ++d) dotv += gs[h * HD + d] * gd[h * HD + d];
      float et = 0.f;
      #pragma unroll
      for (int c = 0; c < EDGE_DIM; ++c) et += eav[c] * sWec[h * EDGE_DIM + c];
      float a = dotv + 8.f * et;              // edge_term = ea * D
      a = (a > 0.f) ? a : 0.2f * a;           // leaky_relu(0.2)
      alpha[(size_t)e * NH + h] = a;
      atomicMax(&smax[h], encOrd(a));
    }
  }
  __syncthreads();
  if (tid < NH) atomicMax(&hmax[tid], smax[tid]);
}

// ---------------- kernel 4: exp(a - max), per-head sum --------------------
__global__ __launch_bounds__(256) void sumexp_kernel(
    float* __restrict__ alpha, const unsigned* __restrict__ hmax,
    float* __restrict__ hsum) {
  __shared__ float ssum[NH];
  __shared__ float smax[NH];
  const int tid = threadIdx.x;
  if (tid < NH) { ssum[tid] = 0.f; smax[tid] = decOrd(hmax[tid]); }
  __syncthreads();
  const int e = blockIdx.x * blockDim.x + tid;
  if (e < N_EDGES) {
    #pragma unroll
    for (int h = 0; h < NH; ++h) {
      float a = alpha[(size_t)e * NH + h];
      float ex = __expf(a - smax[h]);
      alpha[(size_t)e * NH + h] = ex;
      atomicAdd(&ssum[h], ex);
    }
  }
  __syncthreads();
  if (tid < NH) atomicAdd(&hsum[tid], ssum[tid]);
}

// ---------------- kernel 5: scatter s[n,h] += alpha_exp[e,h] at dst -------
__global__ __launch_bounds__(256) void scatter_kernel(
    const float* __restrict__ alpha, const int* __restrict__ dst,
    float* __restrict__ sacc) {
  const int e = blockIdx.x * blockDim.x + threadIdx.x;
  if (e >= N_EDGES) return;
  const int dI = dst[e];
  #pragma unroll
  for (int h = 0; h < NH; ++h)
    atomicAdd(&sacc[(size_t)dI * NH + h], alpha[(size_t)e * NH + h]);
}

// ---------------- kernel 6: out = relu((g * s_norm) @ W_out^T) ------------
__global__ __launch_bounds__(256) void out_kernel(
    const float* __restrict__ g, const float* __restrict__ sacc,
    const float* __restrict__ hsum, const float* __restrict__ W_out,
    float* __restrict__ out) {
  __shared__ float sWout[HD * HDIM];  // [8 x 64]
  __shared__ float sInvSum[NH];
  const int tid = threadIdx.x;
  for (int i = tid; i < HD * HDIM; i += blockDim.x) sWout[i] = W_out[i];
  if (tid < NH) sInvSum[tid] = 1.0f / hsum[tid];
  __syncthreads();
  const int n = blockIdx.x * blockDim.x + tid;
  if (n >= N_NODES) return;
  float agg[HDIM];
  const float* grow = g + (size_t)n * HDIM;
  #pragma unroll
  for (int h = 0; h < NH; ++h) {
    float w = sacc[(size_t)n * NH + h] * sInvSum[h];
    #pragma unroll
    for (int d = 0; d < HD; ++d) agg[h * HD + d] = grow[h * HD + d] * w;
  }
  #pragma unroll
  for (int j = 0; j < HD; ++j) {
    float acc = 0.f;
    #pragma unroll
    for (int i = 0; i < HDIM; ++i) acc += agg[i] * sWout[j * HDIM + i];
    out[(size_t)n * HD + j] = acc > 0.f ? acc : 0.f;  // relu
  }
}

extern "C" void kernel_launch(void* const* d_in, const int* in_sizes, int n_in,
                              void* d_out, int out_size, void* d_ws, size_t ws_size,
                              hipStream_t stream) {
  const float* x          = (const float*)d_in[0];  // [N,128]
  const float* edge_attr  = (const float*)d_in[1];  // [E,16]
  const float* W          = (const float*)d_in[2];  // [64,128]
  const float* W_edge     = (const float*)d_in[3];  // [64,16]
  const float* W_edge_att = (const float*)d_in[4];  // [8,64]
  const float* W_att      = (const float*)d_in[5];  // [8,8]
  const float* W_out      = (const float*)d_in[6];  // [8,64]
  const int*   edge_index = (const int*)d_in[7];    // [2,E]
  const int* src = edge_index;
  const int* dst = edge_index + N_EDGES;
  float* out = (float*)d_out;                       // [N,8]

  // workspace layout (floats)
  float*    Wc    = (float*)d_ws;                     // 8192
  float*    Wec   = Wc + 8192;                        // 128
  float*    g     = Wec + 128;                        // N*64
  float*    alpha = g + (size_t)N_NODES * HDIM;       // E*8
  unsigned* hmax  = (unsigned*)(alpha + (size_t)N_EDGES * NH);  // 8
  float*    hsum  = (float*)(hmax + NH);              // 8
  float*    sacc  = hsum + NH;                        // N*8

  prep_kernel<<<1, 256, 0, stream>>>(W, W_att, W_edge, W_edge_att, Wc, Wec);

  {
    int total = N_NODES * NH;
    init_kernel<<<(total + 255) / 256, 256, 0, stream>>>(hmax, hsum, sacc);
  }
  {
    const int waves = N_NODES / 16;           // 3125 (N divisible by 16)
    const int wavesPerBlk = 8;                // 256 threads
    const int blocks = (waves + wavesPerBlk - 1) / wavesPerBlk;
    g_gemm_kernel<<<blocks, 256, 0, stream>>>(x, Wc, g);
  }
  {
    const int blocks = (N_EDGES + 255) / 256;
    alpha_kernel<<<blocks, 256, 0, stream>>>(g, edge_attr, Wec, src, dst, alpha, hmax);
    sumexp_kernel<<<blocks, 256, 0, stream>>>(alpha, hmax, hsum);
    scatter_kernel<<<blocks, 256, 0, stream>>>(alpha, dst, sacc);
  }
  {
    const int blocks = (N_NODES + 255) / 256;
    out_kernel<<<blocks, 256, 0, stream>>>(g, sacc, hsum, W_out, out);
  }
  (void)in_sizes; (void)n_in; (void)out_size; (void)ws_size;
}